// EdgeAsAttendeeSelfAttention_38362647888359
// MI455X (gfx1250) — compile-verified
//
#include <hip/hip_runtime.h>
#include <hip/hip_bf16.h>
#include <math.h>

// Problem constants (from the reference)
#define BATCH     4
#define N_NODES   512
#define HIDDEN    768
#define NUM_HEADS 12
#define HEAD_DIM  64
#define DEGREE    64
#define MROWS     (BATCH * N_NODES)   // 2048
#define NSEG      (BATCH * N_NODES)   // 2048

// GEMM tiling
#define BM 64          // output rows per workgroup
#define BN 128         // output cols per workgroup
#define BK 32          // k-chunk staged in LDS
#define LDSW 36        // LDS row stride (floats): %4==0 for float4 stores,
                       // 36*r mod 64 spans 16 distinct banks for fragment reads

typedef __attribute__((ext_vector_type(2))) float v2f;
typedef __attribute__((ext_vector_type(4))) float v4f;
typedef __attribute__((ext_vector_type(8))) float v8f;

// ---------------------------------------------------------------------------
// Kernel 1: fused QKV projection GEMM using CDNA5 fp32 WMMA (16x16x4).
//   Out[m,n] = sum_k X[m,k] * W[n,k] + bias[n]   (torch Linear: x @ W.T + b)
//
// 256 threads (8 waves) per workgroup -> one 64x128 output tile of one of
// {Q,K,V} (blockIdx.z). Per 32-wide k-chunk: stage A(64x32) and B(128x32)
// in LDS with coalesced float4 loads, then each wave computes a 16x64 strip:
// 8 k-steps x 4 n-subtiles, A-fragment reused across the 4 WMMAs.
//
// Fragment layouts (fp32 16x16x4):
//  A: lanes 0-15 hold row m=lane, K={k,k+1}; lanes 16-31 row m=lane-16, K={k+2,k+3}
//  B: mirrored K split across half-waves, N = lane&15
//  C/D: VGPR r -> rows r (lanes 0-15) and r+8 (lanes 16-31), N = lane&15
// ---------------------------------------------------------------------------
__global__ __launch_bounds__(256) void qkv_wmma_kernel(
    const float* __restrict__ X,
    const float* __restrict__ Wq, const float* __restrict__ Wk, const float* __restrict__ Wv,
    const float* __restrict__ bq, const float* __restrict__ bk, const float* __restrict__ bv,
    float* __restrict__ Qo, float* __restrict__ Ko, float* __restrict__ Vo)
{
    __shared__ float As[BM * LDSW];
    __shared__ float Bs[BN * LDSW];

    const int tid  = threadIdx.x;
    const int lane = tid & 31;
    const int wave = tid >> 5;     // 0..7
    const int half = lane >> 4;    // 0: K={k,k+1}, 1: K={k+2,k+3}
    const int lm   = lane & 15;
    const int wm   = wave & 3;     // M sub-strip  (4 x 16 rows)
    const int wn   = wave >> 2;    // N half       (2 x 64 cols)

    const int n0 = blockIdx.x * BN;
    const int m0 = blockIdx.y * BM;

    const float* W;
    const float* bias;
    float* Out;
    if (blockIdx.z == 0)      { W = Wq; bias = bq; Out = Qo; }
    else if (blockIdx.z == 1) { W = Wk; bias = bk; Out = Ko; }
    else                      { W = Wv; bias = bv; Out = Vo; }

    // Cooperative stage-in indices: 256 threads, float4 granularity.
    const int ldr = tid >> 3;          // 0..31 (row within a 32-row group)
    const int ldc = (tid & 7) * 4;     // 0,4,...,28 (col within k-chunk)

    v8f acc[4] = {};

    for (int kc = 0; kc < HIDDEN; kc += BK) {
        __syncthreads();   // previous chunk's fragment reads complete

        // A tile: 64 rows x 32 cols  (2 float4 per thread)
        {
            v4f a0 = *(const v4f*)(X + (size_t)(m0 + ldr)      * HIDDEN + kc + ldc);
            v4f a1 = *(const v4f*)(X + (size_t)(m0 + 32 + ldr) * HIDDEN + kc + ldc);
            *(v4f*)&As[ldr * LDSW + ldc]        = a0;
            *(v4f*)&As[(32 + ldr) * LDSW + ldc] = a1;
        }
        // B tile: 128 rows x 32 cols  (4 float4 per thread)
        #pragma unroll
        for (int i = 0; i < 4; ++i) {
            const int r = ldr + 32 * i;
            v4f b = *(const v4f*)(W + (size_t)(n0 + r) * HIDDEN + kc + ldc);
            *(v4f*)&Bs[r * LDSW + ldc] = b;
        }
        __syncthreads();

#if __has_builtin(__builtin_amdgcn_wmma_f32_16x16x4_f32)
        #pragma unroll
        for (int s = 0; s < BK / 4; ++s) {
            const int ks = s * 4 + half * 2;
            v2f a = *(const v2f*)&As[(wm * 16 + lm) * LDSW + ks];
            #pragma unroll
            for (int nn = 0; nn < 4; ++nn) {
                v2f b = *(const v2f*)&Bs[(wn * 64 + nn * 16 + lm) * LDSW + ks];
                acc[nn] = __builtin_amdgcn_wmma_f32_16x16x4_f32(
                    /*neg_a=*/false, a, /*neg_b=*/false, b,
                    /*c_mod=*/(short)0, acc[nn], /*reuse_a=*/false, /*reuse_b=*/false);
            }
        }
#else
        // Scalar fallback matching the WMMA C/D lane layout (compile safety net).
        for (int nn = 0; nn < 4; ++nn)
            for (int r = 0; r < 8; ++r) {
                float s = 0.f;
                for (int k = 0; k < BK; ++k)
                    s += As[(wm * 16 + r + half * 8) * LDSW + k] *
                         Bs[(wn * 64 + nn * 16 + lm) * LDSW + k];
                acc[nn][r] += s;
            }
#endif
    }

    // Epilogue: bias add + store. Row m = m0 + wm*16 + r + half*8,
    // col n = n0 + wn*64 + nn*16 + lm (contiguous across lanes 0-15).
    #pragma unroll
    for (int nn = 0; nn < 4; ++nn) {
        const int n  = n0 + wn * 64 + nn * 16 + lm;
        const float bn = bias[n];
        #pragma unroll
        for (int r = 0; r < 8; ++r) {
            const int m = m0 + wm * 16 + r + half * 8;
            Out[(size_t)m * HIDDEN + n] = acc[nn][r] + bn;
        }
    }
}

// ---------------------------------------------------------------------------
// Kernel 2: per-segment edge attention.
// One 256-thread workgroup (8 waves) per (batch, head_node) segment.
//  - stage Q row, neighbor/tail ids, relation ids, and logits in LDS
//  - per-head serial softmax over the 64 neighbors
//  - weighted scatter-sum into the 768-wide output row
// Neighbor K/V rows are gathered from global; adjacent segments share 63/64
// of their windows so this traffic is L2-resident (12 MB total K+V).
// ---------------------------------------------------------------------------
__global__ __launch_bounds__(256) void edge_attn_kernel(
    const float* __restrict__ Q, const float* __restrict__ K, const float* __restrict__ V,
    const int* __restrict__ edge_indices,   // rows: b, h, t, rel; each length E
    const float* __restrict__ E_key, const float* __restrict__ E_val,
    float* __restrict__ out, int E)
{
    const int seg  = blockIdx.x;            // 0..2047  (b*N_NODES + node)
    const int b    = seg / N_NODES;
    const int tid  = threadIdx.x;

    __shared__ float q[HIDDEN];
    __shared__ float logits[DEGREE * NUM_HEADS];   // [j*NUM_HEADS + h]
    __shared__ float att[DEGREE * NUM_HEADS];
    __shared__ int   tnode[DEGREE];
    __shared__ int   rels[DEGREE];

    for (int i = tid; i < HIDDEN; i += 256)
        q[i] = Q[(size_t)seg * HIDDEN + i];

    const size_t ebase = (size_t)seg * DEGREE;
    if (tid < DEGREE) {
        tnode[tid] = edge_indices[(size_t)2 * E + ebase + tid];   // tail node
        rels[tid]  = edge_indices[(size_t)3 * E + ebase + tid];   // relation id
    }
    __syncthreads();

    // Logits: 64 neighbors x 12 heads = 768 dot products of length 64.
    const float scale = 0.125f;   // 1/sqrt(64)
    for (int idx = tid; idx < DEGREE * NUM_HEADS; idx += 256) {
        const int j = idx / NUM_HEADS;
        const int h = idx % NUM_HEADS;
        const float* __restrict__ krow =
            K + ((size_t)b * N_NODES + tnode[j]) * HIDDEN + h * HEAD_DIM;
        const float* __restrict__ ekrow =
            E_key + (size_t)rels[j] * HIDDEN + h * HEAD_DIM;
        const float* __restrict__ qh = q + h * HEAD_DIM;
        float s = 0.f;
        #pragma unroll 8
        for (int d = 0; d < HEAD_DIM; ++d)
            s = fmaf(qh[d], krow[d] + ekrow[d], s);
        logits[idx] = s * scale;
    }
    __syncthreads();

    // Segment softmax, one head per thread (12 active threads, 64 values each).
    if (tid < NUM_HEADS) {
        float m = -INFINITY;
        for (int j = 0; j < DEGREE; ++j)
            m = fmaxf(m, logits[j * NUM_HEADS + tid]);
        float sum = 0.f;
        for (int j = 0; j < DEGREE; ++j) {
            float e = __expf(logits[j * NUM_HEADS + tid] - m);
            att[j * NUM_HEADS + tid] = e;
            sum += e;
        }
        const float inv = 1.f / sum;
        for (int j = 0; j < DEGREE; ++j)
            att[j * NUM_HEADS + tid] *= inv;
    }
    __syncthreads();

    // Output row: 768 elements, each a 64-term weighted sum over neighbors.
    for (int idx = tid; idx < HIDDEN; idx += 256) {
        const int h = idx / HEAD_DIM;
        float s = 0.f;
        #pragma unroll 4
        for (int j = 0; j < DEGREE; ++j) {
            const float v = V[((size_t)b * N_NODES + tnode[j]) * HIDDEN + idx]
                          + E_val[(size_t)rels[j] * HIDDEN + idx];
            s = fmaf(att[j * NUM_HEADS + h], v, s);
        }
        out[(size_t)seg * HIDDEN + idx] = s;
    }
}

// ---------------------------------------------------------------------------
// Launch
// ---------------------------------------------------------------------------
extern "C" void kernel_launch(void* const* d_in, const int* in_sizes, int n_in,
                              void* d_out, int out_size, void* d_ws, size_t ws_size,
                              hipStream_t stream) {
    const float* node_states  = (const float*)d_in[0];
    const int*   edge_indices = (const int*)d_in[1];
    const float* Wq = (const float*)d_in[2];
    const float* bq = (const float*)d_in[3];
    const float* Wk = (const float*)d_in[4];
    const float* bk = (const float*)d_in[5];
    const float* Wv = (const float*)d_in[6];
    const float* bv = (const float*)d_in[7];
    const float* E_key = (const float*)d_in[8];
    const float* E_val = (const float*)d_in[9];

    const int E = in_sizes[1] / 4;   // edge_indices is (4, E)

    // Workspace layout: Q | K | V, each (2048 x 768) fp32 = 6 MB.
    float* Qb = (float*)d_ws;
    float* Kb = Qb + (size_t)MROWS * HIDDEN;
    float* Vb = Kb + (size_t)MROWS * HIDDEN;

    // Phase 1: QKV projections. 6 N-tiles x 32 M-tiles x 3 matrices,
    // 256 threads per 64x128 tile.
    dim3 gemm_grid(HIDDEN / BN, MROWS / BM, 3);
    qkv_wmma_kernel<<<gemm_grid, 256, 0, stream>>>(
        node_states, Wq, Wk, Wv, bq, bk, bv, Qb, Kb, Vb);

    // Phase 2: per-segment edge attention.
    edge_attn_kernel<<<NSEG, 256, 0, stream>>>(
        Qb, Kb, Vb, edge_indices, E_key, E_val, (float*)d_out, E);
}